// LSTMPolicyARS_58136677318825
// MI455X (gfx1250) — compile-verified
//
#include <hip/hip_runtime.h>
#include <math.h>

typedef float v2f __attribute__((ext_vector_type(2)));
typedef float v8f __attribute__((ext_vector_type(8)));

#define LSTRIDE 68   // padded LDS row stride (floats): banks = 4*col + k (mod 64), conflict-free
#define TSTRIDE 20   // per-wave relayout tile stride: banks 20*col spaced by 4, conflict-free
#define WAVES   8    // waves per block (wave32), 16 batch rows per wave -> 128 rows/block

#define L2E 1.4426950408889634f   // log2(e)

__device__ __forceinline__ float rcp_f(float x)  { return __builtin_amdgcn_rcpf(x); }
__device__ __forceinline__ float exp2_f(float x) { return __builtin_amdgcn_exp2f(x); }

__device__ __forceinline__ v8f wmma_f32(v2f a, v2f b, v8f c) {
    // D = A(16x4 f32) * B(4x16 f32) + C(16x16 f32)
    return __builtin_amdgcn_wmma_f32_16x16x4_f32(false, a, false, b, (short)0, c, false, false);
}

// DS-only fence: LDS ops from one wave execute in order; this stops compiler
// reordering and waits DS completion without draining global store counters.
__device__ __forceinline__ void lds_fence_wave() {
    asm volatile("s_wait_dscnt 0x0" ::: "memory");
}
__device__ __forceinline__ void compiler_mem_fence() {
    asm volatile("" ::: "memory");
}

// ---------------------------------------------------------------------------
// Prep: Mf = W_ih @ W_in  (256x64), biasv = W_ih@b_in + b_ih + b_hh  (256)
// ---------------------------------------------------------------------------
__global__ __launch_bounds__(64) void lstm_prep(
    const float* __restrict__ W_in, const float* __restrict__ b_in,
    const float* __restrict__ W_ih, const float* __restrict__ b_ih,
    const float* __restrict__ b_hh,
    float* __restrict__ Mf, float* __restrict__ biasv) {
    const int n = blockIdx.x;   // 0..255 (gate row)
    const int j = threadIdx.x;  // 0..63  (input column)
    float acc = 0.0f;
#pragma unroll 8
    for (int k = 0; k < 64; ++k) acc += W_ih[n * 64 + k] * W_in[k * 64 + j];
    Mf[n * 64 + j] = acc;
    if (j == 0) {
        float bb = b_ih[n] + b_hh[n];
#pragma unroll 8
        for (int k = 0; k < 64; ++k) bb += W_ih[n * 64 + k] * b_in[k];
        biasv[n] = bb;
    }
}

// ---------------------------------------------------------------------------
// Main: gates = obs@Mf^T + h0@W_hh^T + biasv ; LSTM cell ; action GEMM
// ---------------------------------------------------------------------------
__global__ __launch_bounds__(256) void lstm_main(
    const float* __restrict__ obs, const float* __restrict__ h0,
    const float* __restrict__ c0,
    const float* __restrict__ Mf, const float* __restrict__ biasv,
    const float* __restrict__ Whh, const float* __restrict__ Wout,
    const float* __restrict__ bout,
    float* __restrict__ outA, float* __restrict__ outH,
    float* __restrict__ outC) {
    __shared__ float sM[256 * LSTRIDE];             // fused input->gate weights
    __shared__ float sW[256 * LSTRIDE];             // W_hh
    __shared__ float sWo[16 * LSTRIDE];             // W_out
    __shared__ float sB[256];                       // fused gate bias
    __shared__ float sBo[16];                       // b_out
    __shared__ float sT[WAVES][16 * TSTRIDE];       // per-wave h_new relayout tile

    const int tid   = threadIdx.x;
    const int wave  = tid >> 5;
    const int lane  = tid & 31;
    const int col   = lane & 15;       // N (or M for A-operand rows)
    const int hi    = lane >> 4;       // lane half
    const int khalf = hi << 1;         // K sub-offset for f32 WMMA operands
    const int rbase = hi << 3;         // C/D row base for this lane half
    const long m0   = ((long)blockIdx.x * WAVES + wave) * 16;

    // ---- preload A operands first (overlaps with staging latency) ----
    v2f aO[16], aH[16];
    {
        const float* obsRow = obs + (m0 + col) * 64 + khalf;
        const float* hRow   = h0  + (m0 + col) * 64 + khalf;
#pragma unroll
        for (int kk = 0; kk < 16; ++kk) {
            aO[kk] = *(const v2f*)(obsRow + kk * 4);
            aH[kk] = *(const v2f*)(hRow   + kk * 4);
        }
    }

    // ---- stage weights into LDS (padded rows) ----
    for (int idx = tid; idx < 256 * 64; idx += 256) {
        const int r = idx >> 6, c = idx & 63;
        sM[r * LSTRIDE + c] = Mf[idx];
        sW[r * LSTRIDE + c] = Whh[idx];
    }
    for (int idx = tid; idx < 16 * 64; idx += 256) {
        const int r = idx >> 6, c = idx & 63;
        sWo[r * LSTRIDE + c] = Wout[idx];
    }
    sB[tid] = biasv[tid];
    if (tid < 16) sBo[tid] = bout[tid];
    __syncthreads();   // only block-wide barrier in the kernel

    float* myT = &sT[wave][0];
    v8f accA = {};                  // action accumulator; bias folded into tanh
    const float boutl = sBo[col];

    for (int j = 0; j < 4; ++j) {
        const int nI = (j * 16 + col) * LSTRIDE + khalf;          // i-gate rows
        const float* bMi  = &sM[nI];
        const float* bMf  = &sM[nI +  64 * LSTRIDE];
        const float* bMg  = &sM[nI + 128 * LSTRIDE];
        const float* bMo  = &sM[nI + 192 * LSTRIDE];
        const float* bWi  = &sW[nI];
        const float* bWf  = &sW[nI +  64 * LSTRIDE];
        const float* bWg  = &sW[nI + 128 * LSTRIDE];
        const float* bWo2 = &sW[nI + 192 * LSTRIDE];

        const int hcol = j * 16 + col;

        // preload c0 for this chunk (consumed after 128 WMMAs)
        float cv[8];
#pragma unroll
        for (int r = 0; r < 8; ++r) cv[r] = c0[(m0 + rbase + r) * 64 + hcol];

        // exp2-domain bias constants (bias folded into activation FMA)
        const float cbi = -L2E * sB[j * 16 + col];
        const float cbf = -L2E * sB[64 + j * 16 + col];
        const float cbg = (2.0f * L2E) * sB[128 + j * 16 + col];
        const float cbo = -L2E * sB[192 + j * 16 + col];

        v8f ai = {}, af = {}, ag = {}, ao = {};   // inline-0 C on first WMMA

#pragma unroll
        for (int kk = 0; kk < 16; ++kk) {
            const int o4 = kk * 4;
            v2f mi = *(const v2f*)(bMi + o4);
            v2f mf = *(const v2f*)(bMf + o4);
            v2f mg = *(const v2f*)(bMg + o4);
            v2f mo = *(const v2f*)(bMo + o4);
            ai = wmma_f32(aO[kk], mi, ai);
            af = wmma_f32(aO[kk], mf, af);
            ag = wmma_f32(aO[kk], mg, ag);
            ao = wmma_f32(aO[kk], mo, ao);
            v2f wi = *(const v2f*)(bWi + o4);
            v2f wf = *(const v2f*)(bWf + o4);
            v2f wg = *(const v2f*)(bWg + o4);
            v2f wo = *(const v2f*)(bWo2 + o4);
            ai = wmma_f32(aH[kk], wi, ai);
            af = wmma_f32(aH[kk], wf, af);
            ag = wmma_f32(aH[kk], wg, ag);
            ao = wmma_f32(aH[kk], wo, ao);
        }

        // ---- LSTM cell elementwise on this 16x16 H-chunk (native exp2/rcp) ----
#pragma unroll
        for (int r = 0; r < 8; ++r) {
            const long row = m0 + rbase + r;
            // sigmoid(x+b) = rcp(1 + 2^(-L*x - L*b))
            const float iv = rcp_f(1.0f + exp2_f(fmaf(ai[r], -L2E, cbi)));
            const float fv = rcp_f(1.0f + exp2_f(fmaf(af[r], -L2E, cbf)));
            const float ov = rcp_f(1.0f + exp2_f(fmaf(ao[r], -L2E, cbo)));
            // tanh(x+b) = 1 - 2*rcp(1 + 2^(2L*x + 2L*b))
            const float gv = 1.0f - 2.0f * rcp_f(1.0f + exp2_f(fmaf(ag[r], 2.0f * L2E, cbg)));
            const float cn = fmaf(fv, cv[r], iv * gv);
            const float hn = ov * (1.0f - 2.0f * rcp_f(1.0f + exp2_f((2.0f * L2E) * cn)));
            outC[row * 64 + hcol] = cn;
            outH[row * 64 + hcol] = hn;
            myT[(rbase + r) * TSTRIDE + col] = hn;   // D-layout -> LDS for relayout
        }

        // wave-private LDS tile: in-order LDS pipeline guarantees cross-lane RAW;
        // fence only stops compiler reordering + waits DS (not global stores).
        lds_fence_wave();

        // ---- partial action GEMM: accA += h_new_chunk @ W_out[:, j*16:+16]^T
        const float* aT = myT + col * TSTRIDE + khalf;            // A: h_new rows
        const float* bO = &sWo[col * LSTRIDE + j * 16 + khalf];   // B: W_out rows
#pragma unroll
        for (int kk = 0; kk < 4; ++kk) {
            v2f a = *(const v2f*)(aT + kk * 4);
            v2f b = *(const v2f*)(bO + kk * 4);
            accA = wmma_f32(a, b, accA);
        }
        compiler_mem_fence();   // WAR vs next chunk's sT stores (HW is in-order)
    }

    // ---- action = tanh(acc + b_out); [-1,1] range maps to itself ----
    const float cba = (2.0f * L2E) * boutl;
#pragma unroll
    for (int r = 0; r < 8; ++r) {
        const long row = m0 + rbase + r;
        outA[row * 16 + col] =
            1.0f - 2.0f * rcp_f(1.0f + exp2_f(fmaf(accA[r], 2.0f * L2E, cba)));
    }
}

extern "C" void kernel_launch(void* const* d_in, const int* in_sizes, int n_in,
                              void* d_out, int out_size, void* d_ws, size_t ws_size,
                              hipStream_t stream) {
    const float* obs   = (const float*)d_in[0];
    const float* h0    = (const float*)d_in[1];
    const float* c0    = (const float*)d_in[2];
    const float* W_in  = (const float*)d_in[3];
    const float* b_in  = (const float*)d_in[4];
    const float* W_ih  = (const float*)d_in[5];
    const float* W_hh  = (const float*)d_in[6];
    const float* b_ih  = (const float*)d_in[7];
    const float* b_hh  = (const float*)d_in[8];
    const float* W_out = (const float*)d_in[9];
    const float* b_out = (const float*)d_in[10];

    const int B = in_sizes[0] / 64;          // 262144
    float* out = (float*)d_out;
    float* Mf    = (float*)d_ws;             // 256*64 floats
    float* biasv = Mf + 256 * 64;            // 256 floats

    lstm_prep<<<256, 64, 0, stream>>>(W_in, b_in, W_ih, b_ih, b_hh, Mf, biasv);

    float* outA = out;                           // [B,16]
    float* outH = out + (size_t)B * 16;          // [1,B,64]
    float* outC = out + (size_t)B * 80;          // [1,B,64]
    lstm_main<<<B / (16 * WAVES), 256, 0, stream>>>(
        obs, h0, c0, Mf, biasv, W_hh, W_out, b_out, outA, outH, outC);
}